// RepulsionCalculator_12438225289630
// MI455X (gfx1250) — compile-verified
//
#include <hip/hip_runtime.h>
#include <stdint.h>

// ---------------------------------------------------------------------------
// Repulsion energy: rep = (y/d) * exp(-sa * d^k) * exp(1 - 1/(1 - (d/rc)^2))
// segment-summed per molecule.
//   Pass 1 (bandwidth-bound): async Global->LDS species staging, B128 pair
//           streaming, ds_add_f32 256-bin block reduction, partials -> ws.
//   Pass 2: WMMA f32 16x16x4 row-reduction of the (blocks x bins) partial
//           matrix with an all-ones A matrix (tensor-core reduction).
// ---------------------------------------------------------------------------

#define ANG2BOHR 1.8897261258369282f
#define RCUT     (5.2f * ANG2BOHR)
#define INV_RCUT (1.0f / (5.2f * ANG2BOHR))
#define NELEM    4
#define NBINS    256
#define MAX_SPECIES 16384   // B*A for the reference problem (64 KB in LDS)

typedef __attribute__((ext_vector_type(2))) float v2f;
typedef __attribute__((ext_vector_type(8))) float v8f;

// CDNA5 async global->LDS copy (128b per lane), tracked by ASYNCcnt.
__device__ __forceinline__ void async_ld_b128(uint32_t lds_off, const void* gptr) {
    asm volatile("global_load_async_to_lds_b128 %0, %1, off"
                 :: "v"(lds_off), "v"(gptr)
                 : "memory");
}
__device__ __forceinline__ void wait_asynccnt0() {
    asm volatile("s_wait_asynccnt 0" ::: "memory");
}

__global__ void repulsion_init_out(const float* __restrict__ energies,
                                   float* __restrict__ out, int n) {
    int i = blockIdx.x * blockDim.x + threadIdx.x;
    if (i < n) out[i] = energies[i];
}

__device__ __forceinline__ void do_pair(int i0, int i1, float dist_ang,
                                        const int* __restrict__ s_sp,
                                        const float4* __restrict__ s_tab,
                                        float* __restrict__ s_bins,
                                        int mol_shift, int num_atoms) {
    float d = dist_ang * ANG2BOHR;
    int combo = s_sp[i0] * NELEM + s_sp[i1];
    float4 t = s_tab[combo];            // {y, sqrt_alpha, k, 0}
    // smooth cutoff argument: 1 - 1/(1-x^2), x = clamp(d/rc, 0, 1-1e-6)
    float x  = fminf(d * INV_RCUT, 1.0f - 1e-6f);   // d > 0 so no lower clamp
    float om = 1.0f - x * x;
    float cut_arg = 1.0f - __builtin_amdgcn_rcpf(om);
    // d^k = exp(k * log d); fold both exponentials into one v_exp_f32
    float powdk = __expf(t.z * __logf(d));
    float e     = __expf(cut_arg - t.y * powdk);
    float rep   = t.x * __builtin_amdgcn_rcpf(d) * e;
    rep = (d < RCUT) ? rep : 0.0f;
    int mol = (mol_shift >= 0) ? (i0 >> mol_shift) : (i0 / num_atoms);
    atomicAdd(&s_bins[mol], rep);       // ds_add_f32 (no return)
}

__global__ __launch_bounds__(256) void repulsion_kernel(
    const int*   __restrict__ species, int n_species,
    const int*   __restrict__ idx0,
    const int*   __restrict__ idx1,
    const float* __restrict__ dist, int P,
    const float* __restrict__ y_ab,
    const float* __restrict__ sa_ab,
    const float* __restrict__ k_ab,
    float*       __restrict__ out, int out_n,
    float*       __restrict__ ws, int ws_mode,
    int mol_shift, int num_atoms)
{
    __shared__ __align__(16) int    s_sp[MAX_SPECIES];   // 64 KB
    __shared__ __align__(16) float4 s_tab[NELEM * NELEM];
    __shared__ float s_bins[NBINS];

    // --- stage species table into LDS via CDNA5 async copies ---------------
    {
        // LDS aperture keeps the wave-relative LDS offset in addr[31:0]
        uint32_t lds_base = (uint32_t)(uintptr_t)&s_sp[0];
        int nbytes = n_species * 4;
        if (nbytes > MAX_SPECIES * 4) nbytes = MAX_SPECIES * 4;
        const char* g = (const char*)species;
        for (int off = threadIdx.x * 16; off < nbytes; off += blockDim.x * 16) {
            async_ld_b128(lds_base + (uint32_t)off, g + off);
        }
    }
    if (threadIdx.x < NELEM * NELEM) {
        s_tab[threadIdx.x] = make_float4(y_ab[threadIdx.x], sa_ab[threadIdx.x],
                                         k_ab[threadIdx.x], 0.0f);
    }
    for (int b = threadIdx.x; b < NBINS; b += blockDim.x) s_bins[b] = 0.0f;
    wait_asynccnt0();
    __syncthreads();

    // --- grid-stride loop over 4-pair chunks (B128 vectorized loads) -------
    int tid      = blockIdx.x * blockDim.x + threadIdx.x;
    int nthreads = gridDim.x * blockDim.x;
    int chunks   = P >> 2;
    const int4*   v0 = (const int4*)idx0;
    const int4*   v1 = (const int4*)idx1;
    const float4* vd = (const float4*)dist;

    for (int c = tid; c < chunks; c += nthreads) {
        __builtin_prefetch(&vd[c + nthreads], 0, 0);   // global_prefetch_b8
        int4   a  = v0[c];
        int4   b  = v1[c];
        float4 dd = vd[c];
        do_pair(a.x, b.x, dd.x, s_sp, s_tab, s_bins, mol_shift, num_atoms);
        do_pair(a.y, b.y, dd.y, s_sp, s_tab, s_bins, mol_shift, num_atoms);
        do_pair(a.z, b.z, dd.z, s_sp, s_tab, s_bins, mol_shift, num_atoms);
        do_pair(a.w, b.w, dd.w, s_sp, s_tab, s_bins, mol_shift, num_atoms);
    }
    // tail (P not divisible by 4)
    for (int p = (chunks << 2) + tid; p < P; p += nthreads) {
        do_pair(idx0[p], idx1[p], dist[p], s_sp, s_tab, s_bins, mol_shift, num_atoms);
    }

    __syncthreads();
    if (ws_mode) {
        // store per-block partial sums; reduced by the WMMA kernel
        float* row = ws + (size_t)blockIdx.x * out_n;
        for (int b = threadIdx.x; b < out_n; b += blockDim.x) row[b] = s_bins[b];
    } else {
        int lim = out_n < NBINS ? out_n : NBINS;
        for (int b = threadIdx.x; b < lim; b += blockDim.x) {
            float v = s_bins[b];
            if (v != 0.0f) unsafeAtomicAdd(&out[b], v);  // global_atomic_add_f32
        }
    }
}

// ---------------------------------------------------------------------------
// Row-reduction of ws (rows x cols) via V_WMMA_F32_16X16X4_F32:
//   D(16x16) += ones(16x4) x B(4x16)  =>  every row of D = column sums of the
//   4-row chunk; accumulate over all chunks, then lanes 0..15 extract N=0..15
//   from C/D VGPR0 (M=0 row). Result is K-layout-permutation invariant since
//   A is all ones. One wave per 16 output columns.
// ---------------------------------------------------------------------------
__global__ __launch_bounds__(512) void wmma_reduce(const float* __restrict__ ws,
                                                   float* __restrict__ out,
                                                   int rows, int cols) {
    int wave = (blockIdx.x * blockDim.x + threadIdx.x) >> 5;
    int lane = threadIdx.x & 31;
    int col0 = wave * 16;
    if (col0 >= cols) return;              // wave-uniform: EXEC stays all-ones
    int cL   = lane & 15;
    int rsel = (lane >> 4) << 1;           // lanes 0-15 -> rows K=0,1; 16-31 -> K=2,3
    const float* base = ws + col0 + cL;

    v8f acc = {};
    v2f a;  a.x = 1.0f; a.y = 1.0f;        // all-ones A matrix
    for (int r = 0; r < rows; r += 4) {
        v2f b;
        b.x = base[(size_t)(r + rsel)     * cols];
        b.y = base[(size_t)(r + rsel + 1) * cols];
        acc = __builtin_amdgcn_wmma_f32_16x16x4_f32(
                  false, a, false, b, (short)0, acc, false, false);
    }
    if (lane < 16) out[col0 + cL] += acc[0];   // single writer per column
}

extern "C" void kernel_launch(void* const* d_in, const int* in_sizes, int n_in,
                              void* d_out, int out_size, void* d_ws, size_t ws_size,
                              hipStream_t stream) {
    const int*   species  = (const int*)  d_in[0];
    const float* energies = (const float*)d_in[1];
    const int*   ai12     = (const int*)  d_in[2];
    const float* dist     = (const float*)d_in[3];
    const float* y_ab     = (const float*)d_in[4];
    const float* sa_ab    = (const float*)d_in[5];
    const float* k_ab     = (const float*)d_in[6];

    int n_species = in_sizes[0];
    int P         = in_sizes[3];
    float* out    = (float*)d_out;
    float* ws     = (float*)d_ws;

    int num_atoms = (out_size > 0) ? (n_species / out_size) : 1;
    if (num_atoms <= 0) num_atoms = 1;
    int mol_shift = -1;
    if ((num_atoms & (num_atoms - 1)) == 0) mol_shift = __builtin_ctz(num_atoms);

    // out = energies (reference accumulates into the input energies)
    repulsion_init_out<<<(out_size + 255) / 256, 256, 0, stream>>>(energies, out, out_size);

    // 1024 blocks x 256 threads (8 wave32s), ~65 KB LDS -> up to 4 WGs/WGP;
    // each block-iteration amortizes the 64 KB species staging over ~8K pairs.
    const int MAIN_BLOCKS = 1024;   // multiple of 4 (WMMA K-chunking)
    int ws_mode = (out_size > 0) && (out_size <= NBINS) && (out_size % 16 == 0) &&
                  ((size_t)MAIN_BLOCKS * (size_t)out_size * sizeof(float) <= ws_size);

    repulsion_kernel<<<MAIN_BLOCKS, 256, 0, stream>>>(
        species, n_species, ai12, ai12 + P, dist, P,
        y_ab, sa_ab, k_ab, out, out_size,
        ws, ws_mode, mol_shift, num_atoms);

    if (ws_mode) {
        int waves  = out_size / 16;                // one wave per 16 columns
        int blocks = (waves * 32 + 511) / 512;
        wmma_reduce<<<blocks, 512, 0, stream>>>(ws, out, MAIN_BLOCKS, out_size);
    }
}